// S4DTrace_11708080849234
// MI455X (gfx1250) — compile-verified
//
#include <hip/hip_runtime.h>
#include <hip/hip_bf16.h>

// ---------------------------------------------------------------------------
// S4D linear recurrence on MI455X (gfx1250), wave32, WMMA f32 16x16x4.
//
//   w_t = G * [w_{t-1}; x_t],  G : 128x192 real (from M = V diag(A),
//   C = V diag(dt) Vinv),  y_t = 2 * w_t[0:64].
//
// Chunked parallel scan: pass A (local scan, zero init) -> combine
// (M^L by squaring + sequential chunk-prefix) -> pass C (homogeneous
// correction added to output).  8 chain-groups x NC chunks workgroups for
// the heavy passes; each WG = 4 waves, 16 chains, f32 WMMA per step.
// NC picked at runtime from ws_size (deterministic): 32 if it fits
// (256 WGs, L=64), else 16, 8, ... 1.
// ---------------------------------------------------------------------------

typedef float v2f __attribute__((ext_vector_type(2)));
typedef float v8f __attribute__((ext_vector_type(8)));

#define T_LEN   2048
#define B_DIM   8
#define NFEAT   1024
#define H_ST    64          // complex state dim
#define W_ST    128         // real lifted state dim
#define K_FULL  192         // [w ; x]
#define GROUPS  8           // chain groups (== B), 16 chains each
#define PAD     20          // LDS row stride (floats), bank-conflict padding

// ---- workspace layout (floats) ----
// G      : 128*192          = 24576   @ 0
// P=M^L  : 128*128          = 16384   @ 24576
// small  : dt,Ar,Ai (3*64)  = 192     @ 40960
// r      : NC*8*128*16                @ 41152
// s      : NC*8*128*16                @ 41152 + NC*8*2048
#define OFF_P     24576
#define OFF_SMALL 40960
#define OFF_R     41152

static __device__ inline v8f wmma4(v2f a, v2f b, v8f c) {
    // D = A(16x4 f32) * B(4x16 f32) + C(16x16 f32)
    return __builtin_amdgcn_wmma_f32_16x16x4_f32(
        false, a, false, b, (short)0, c, false, false);
}

// --- prep 1: dt, A = exp((Lr + i Li) dt) ---------------------------------
__global__ void prep_small_k(const float* __restrict__ logdt,
                             const float* __restrict__ lLr,
                             const float* __restrict__ Li,
                             float* __restrict__ small) {
    int h = threadIdx.x;
    if (h < H_ST) {
        float dt  = expf(logdt[h]);
        float lr  = -expf(lLr[h]);
        float mag = expf(lr * dt);
        float th  = Li[h] * dt;
        small[h]            = dt;
        small[H_ST + h]     = mag * cosf(th);   // Ar
        small[2 * H_ST + h] = mag * sinf(th);   // Ai
    }
}

// --- prep 2: build G = [[Mr,-Mi,Cr],[Mi,Mr,Ci]] (row-major 128x192) ------
__global__ void build_G_k(const float* __restrict__ V,
                          const float* __restrict__ Vinv,
                          const float* __restrict__ small,
                          float* __restrict__ G) {
    int r = blockIdx.x;      // 0..127
    int c = threadIdx.x;     // 0..191
    const float* dt = small;
    const float* Ar = small + H_ST;
    const float* Ai = small + 2 * H_ST;
    bool top = (r < H_ST);
    int  i   = top ? r : r - H_ST;
    float val;
    if (c < W_ST) {
        bool left = (c < H_ST);
        int  j    = left ? c : c - H_ST;
        float vr = V[(i * H_ST + j) * 2];
        float vi = V[(i * H_ST + j) * 2 + 1];
        float mr = vr * Ar[j] - vi * Ai[j];     // Re(V[i,j]*A[j])
        float mi = vr * Ai[j] + vi * Ar[j];     // Im(V[i,j]*A[j])
        val = top ? (left ? mr : -mi) : (left ? mi : mr);
    } else {
        int j = c - W_ST;                        // C[:, j]
        float re = 0.f, im = 0.f;
        for (int k = 0; k < H_ST; ++k) {
            float vr = V[(i * H_ST + k) * 2];
            float vi = V[(i * H_ST + k) * 2 + 1];
            float wr = Vinv[(k * H_ST + j) * 2];
            float wi = Vinv[(k * H_ST + j) * 2 + 1];
            float d  = dt[k];
            re += d * (vr * wr - vi * wi);
            im += d * (vr * wi + vi * wr);
        }
        val = top ? re : im;
    }
    G[r * K_FULL + c] = val;
}

// --- P = Mblk^L by repeated squaring (one WG, dynamic LDS 128KB) ---------
__global__ void matpow_k(const float* __restrict__ G, float* __restrict__ P,
                         int logL) {
    extern __shared__ float lds[];
    float* A  = lds;                   // 128*128
    float* Bm = lds + W_ST * W_ST;     // 128*128
    for (int i = threadIdx.x; i < W_ST * W_ST; i += blockDim.x) {
        int r = i >> 7, c = i & 127;
        A[i] = G[r * K_FULL + c];
    }
    __syncthreads();
    for (int it = 0; it < logL; ++it) {
        for (int i = threadIdx.x; i < W_ST * W_ST; i += blockDim.x) {
            int r = i >> 7, c = i & 127;
            float acc = 0.f;
            for (int k = 0; k < W_ST; ++k) acc += A[r * W_ST + k] * A[k * W_ST + c];
            Bm[i] = acc;
        }
        __syncthreads();
        for (int i = threadIdx.x; i < W_ST * W_ST; i += blockDim.x) A[i] = Bm[i];
        __syncthreads();
    }
    for (int i = threadIdx.x; i < W_ST * W_ST; i += blockDim.x) P[i] = A[i];
}

// --- chunk-prefix combine: s_0 = 0, s_{c+1} = P s_c + r_c ----------------
__global__ void combine_k(const float* __restrict__ P,
                          const float* __restrict__ rws,
                          float* __restrict__ sws, int NC) {
    int g   = blockIdx.x;       // chain group
    int row = threadIdx.x;      // 0..127 state row
    __shared__ float s_lds[W_ST][16];
    for (int n = 0; n < 16; ++n) s_lds[row][n] = 0.f;
    float* s0 = sws + (size_t)g * 2048 + row * 16;
    for (int n = 0; n < 16; ++n) s0[n] = 0.f;
    __syncthreads();
    for (int c = 1; c < NC; ++c) {
        float acc[16];
        for (int n = 0; n < 16; ++n) acc[n] = 0.f;
        const float* Pr = P + row * W_ST;
        for (int k = 0; k < W_ST; ++k) {
            float pk = Pr[k];
            for (int n = 0; n < 16; ++n) acc[n] += pk * s_lds[k][n];
        }
        const float* rr = rws + ((size_t)(c - 1) * GROUPS + g) * 2048 + row * 16;
        for (int n = 0; n < 16; ++n) acc[n] += rr[n];
        __syncthreads();
        float* so = sws + ((size_t)c * GROUPS + g) * 2048 + row * 16;
        for (int n = 0; n < 16; ++n) { s_lds[row][n] = acc[n]; so[n] = acc[n]; }
        __syncthreads();
    }
}

// --- WMMA scan engine -----------------------------------------------------
// KT K-tiles of 4.  Pass A: KT=48 (state+input), zero init, overwrite out,
// store final state.  Pass C: KT=32 (state only), init from s, add to out.
template <int KT, bool HAS_X, bool ADD_OUT, bool STORE_FINAL>
__global__ __launch_bounds__(128)
void scan_engine(const float* __restrict__ G,
                 const float* __restrict__ xs,
                 float* __restrict__ out,
                 const float* __restrict__ sws,
                 float* __restrict__ rws,
                 int L) {
    constexpr int SROW = KT * 4;                 // B-operand rows
    __shared__ float SB[2][SROW][PAD];
    const int g = blockIdx.x;                    // chain group == b
    const int c = blockIdx.y;                    // chunk
    if (!HAS_X && c == 0) return;                // zero correction for chunk 0

    const int tid  = threadIdx.x;
    const int wave = tid >> 5;
    const int lane = tid & 31;
    const int lHi  = lane >> 4;                  // K half
    const int lLo  = lane & 15;                  // M row / N col within tile

    // stationary A operand (G) in registers, exact WMMA A layout:
    // element(m,k): lane = m + 16*(k pair half), vgpr pair = {k, k+1}
    v2f Areg[2][KT];
#pragma unroll
    for (int at = 0; at < 2; ++at) {
        const int mrow = (wave * 2 + at) * 16 + lLo;
        const float* gp = G + mrow * K_FULL + lHi * 2;
#pragma unroll
        for (int kt = 0; kt < KT; ++kt) {
            Areg[at][kt].x = gp[kt * 4 + 0];
            Areg[at][kt].y = gp[kt * 4 + 1];
        }
    }

    // init state rows of SB[0]
    if (HAS_X) {
        for (int idx = tid; idx < W_ST * 16; idx += 128)
            SB[0][idx >> 4][idx & 15] = 0.f;
    } else {
        const float* sp = sws + ((size_t)c * GROUPS + g) * 2048;
        for (int idx = tid; idx < W_ST * 16; idx += 128)
            SB[0][idx >> 4][idx & 15] = sp[idx];
    }
    const int t0 = c * L;
    if (HAS_X) {   // preload x_{t0}: global (p,h) contiguous -> LDS [128+h][p]
        const float4* xb = (const float4*)(xs + (size_t)t0 * (B_DIM * NFEAT)
                                              + g * NFEAT) + tid * 2;
        float4 q0 = xb[0], q1 = xb[1];
#pragma unroll
        for (int e = 0; e < 4; ++e) {
            int i0 = tid * 8 + e, i1 = tid * 8 + 4 + e;
            SB[0][W_ST + (i0 & 63)][i0 >> 6] = ((const float*)&q0)[e];
            SB[0][W_ST + (i1 & 63)][i1 >> 6] = ((const float*)&q1)[e];
        }
    }
    __syncthreads();

    int cur = 0;
    for (int j = 0; j < L; ++j) {
        const int t   = t0 + j;
        const int nxt = cur ^ 1;
        if (HAS_X && (j + 1 < L)) {   // prefetch x_{t+1} into SB[nxt]
            const float4* xb = (const float4*)(xs + (size_t)(t + 1) * (B_DIM * NFEAT)
                                                  + g * NFEAT) + tid * 2;
            float4 q0 = xb[0], q1 = xb[1];
#pragma unroll
            for (int e = 0; e < 4; ++e) {
                int i0 = tid * 8 + e, i1 = tid * 8 + 4 + e;
                SB[nxt][W_ST + (i0 & 63)][i0 >> 6] = ((const float*)&q0)[e];
                SB[nxt][W_ST + (i1 & 63)][i1 >> 6] = ((const float*)&q1)[e];
            }
        }
        v8f acc0 = {0.f, 0.f, 0.f, 0.f, 0.f, 0.f, 0.f, 0.f};
        v8f acc1 = acc0;
#pragma unroll
        for (int kt = 0; kt < KT; ++kt) {
            v2f b;
            const int kr = kt * 4 + lHi * 2;
            b.x = SB[cur][kr][lLo];
            b.y = SB[cur][kr + 1][lLo];
            acc0 = wmma4(Areg[0][kt], b, acc0);
            acc1 = wmma4(Areg[1][kt], b, acc1);
        }
        // D(16x16): vgpr r holds rows (r, r+8) split across lane halves
#pragma unroll
        for (int rr = 0; rr < 8; ++rr) {
            SB[nxt][wave * 32 + rr + 8 * lHi][lLo]      = acc0[rr];
            SB[nxt][wave * 32 + 16 + rr + 8 * lHi][lLo] = acc1[rr];
        }
        __syncthreads();
        // y_t = 2 * w[0:64]; coalesced 4KB b128 stores (feat = p*64 + h)
        float4* ob = (float4*)(out + (size_t)t * (B_DIM * NFEAT) + g * NFEAT) + tid * 2;
        float4 o0, o1;
#pragma unroll
        for (int e = 0; e < 4; ++e) {
            int i0 = tid * 8 + e, i1 = tid * 8 + 4 + e;
            ((float*)&o0)[e] = 2.0f * SB[nxt][i0 & 63][i0 >> 6];
            ((float*)&o1)[e] = 2.0f * SB[nxt][i1 & 63][i1 >> 6];
        }
        if (ADD_OUT) {
            float4 p0 = ob[0], p1 = ob[1];
            o0.x += p0.x; o0.y += p0.y; o0.z += p0.z; o0.w += p0.w;
            o1.x += p1.x; o1.y += p1.y; o1.z += p1.z; o1.w += p1.w;
        }
        ob[0] = o0; ob[1] = o1;
        cur = nxt;
    }
    if (STORE_FINAL) {
        float* rp = rws + ((size_t)c * GROUPS + g) * 2048;
        for (int idx = tid; idx < W_ST * 16; idx += 128)
            rp[idx] = SB[cur][idx >> 4][idx & 15];
    }
}

extern "C" void kernel_launch(void* const* d_in, const int* in_sizes, int n_in,
                              void* d_out, int out_size, void* d_ws, size_t ws_size,
                              hipStream_t stream) {
    (void)in_sizes; (void)n_in; (void)out_size;
    const float* xs    = (const float*)d_in[0];
    const float* logdt = (const float*)d_in[1];
    const float* lLr   = (const float*)d_in[2];
    const float* Li    = (const float*)d_in[3];
    const float* V     = (const float*)d_in[4];   // complex64 interleaved
    const float* Vinv  = (const float*)d_in[5];   // complex64 interleaved
    float* out = (float*)d_out;
    float* ws  = (float*)d_ws;

    // Pick chunk count from available workspace (deterministic: ws_size fixed).
    // r + s need NC*8*2048 floats each.
    int NC = 1;
    for (int cand = 32; cand >= 2; cand >>= 1) {
        size_t need = ((size_t)OFF_R + 2ull * cand * GROUPS * 2048) * sizeof(float);
        if (ws_size >= need) { NC = cand; break; }
    }
    const int L = T_LEN / NC;
    int logL = 0;
    while ((1 << logL) < L) ++logL;

    float* G     = ws;
    float* P     = ws + OFF_P;
    float* small = ws + OFF_SMALL;
    float* rws   = ws + OFF_R;
    float* sws   = rws + (size_t)NC * GROUPS * 2048;

    prep_small_k<<<1, 64, 0, stream>>>(logdt, lLr, Li, small);
    build_G_k<<<W_ST, K_FULL, 0, stream>>>(V, Vinv, small, G);
    matpow_k<<<1, 128, 2 * W_ST * W_ST * sizeof(float), stream>>>(G, P, logL);

    // pass A: local scans (zero init), 8 groups x NC chunks WGs
    scan_engine<48, true, false, true>
        <<<dim3(GROUPS, NC), 128, 0, stream>>>(G, xs, out, nullptr, rws, L);
    // combine chunk prefixes (sequential over NC chunks, per chain group)
    combine_k<<<GROUPS, 128, 0, stream>>>(P, rws, sws, NC);
    // pass C: homogeneous correction added into out
    scan_engine<32, false, true, false>
        <<<dim3(GROUPS, NC), 128, 0, stream>>>(G, nullptr, out, sws, nullptr, L);
}